// SPL_24936580121263
// MI455X (gfx1250) — compile-verified
//
#include <hip/hip_runtime.h>
#include <math.h>

typedef __attribute__((ext_vector_type(2))) float v2f;
typedef __attribute__((ext_vector_type(8))) float v8f;

// Problem constants
#define HW     128
#define IMG    (HW*HW)        // 16384 per image
#define NMAT   8              // 4 pred + 4 tgt images
#define NPIX   65536          // total pixels per tensor
#define NINT   1000

// ws layout (float offsets)
#define WS_CS     0           // 8 * 16384 column-sum matrices
#define WS_PD     131072      // pred_dist (65536)
#define WS_TD     196608      // tgt_dist  (65536)
#define WS_WPART  262144      // 64 floats
#define WS_MNPART 262208      // 64 floats
#define WS_MXPART 262272      // 64 floats
#define WS_SCAL   262336      // [0]=lo [1]=hi [2]=wsum
#define WS_IPART  262400      // 64 ints (aliased)

__device__ __forceinline__ float band9(int r, int c) {
    int d = r - c; if (d < 0) d = -d;
    return (d <= 4) ? 1.0f : 0.0f;
}
__device__ __forceinline__ float rc9(int i) {
    int a = i < 4 ? i : 4;
    int b = (127 - i) < 4 ? (127 - i) : 4;
    return (float)(a + b + 1);
}

// ---- Kernel 1: ColSum[mat] = Band9 x X[mat]   (one wave per 16x16 tile) ----
// Fixed 6 K-steps of 4 starting at max(0, i0-4): straight-line, EXEC untouched.
__global__ void spl_colsum(const float* __restrict__ pred,
                           const float* __restrict__ tgt,
                           float* __restrict__ cs) {
    int wid  = blockIdx.x * (blockDim.x >> 5) + (threadIdx.x >> 5);
    int lane = threadIdx.x & 31;
    int half = lane >> 4, lm = lane & 15;
    int mat = wid >> 6;
    int t   = wid & 63;
    int i0 = (t >> 3) << 4, j0 = (t & 7) << 4;
    const float* X = (mat < 4) ? (pred + mat * IMG) : (tgt + (mat - 4) * IMG);
    float* C = cs + mat * IMG;

    int kbase = i0 - 4; if (kbase < 0) kbase = 0;   // multiple of 4
    int i = i0 + lm;
    v8f acc = {};
#pragma unroll
    for (int it = 0; it < 6; ++it) {
        int ka = kbase + 4 * it + 2 * half;          // even, may reach 130
        int kb = ka + 1;
        v2f a, b;
        a.x = (ka < HW) ? band9(i, ka) : 0.0f;       // zero A kills OOB columns
        a.y = (kb < HW) ? band9(i, kb) : 0.0f;
        int kca = ka < (HW - 1) ? ka : (HW - 1);     // clamp load rows in-bounds
        int kcb = kb < (HW - 1) ? kb : (HW - 1);
        b.x = X[kca * HW + j0 + lm];
        b.y = X[kcb * HW + j0 + lm];
        acc = __builtin_amdgcn_wmma_f32_16x16x4_f32(false, a, false, b,
                                                    (short)0, acc, false, false);
    }
#pragma unroll
    for (int v = 0; v < 8; ++v) {
        int m = i0 + v + 8 * half;
        C[m * HW + j0 + lm] = acc[v];
    }
}

// ---- Kernel 2: Box = ColSum x Band9^T; dist = X - Box/den ----
__global__ void spl_rowsum_dist(const float* __restrict__ pred,
                                const float* __restrict__ tgt,
                                const float* __restrict__ cs,
                                float* __restrict__ ws) {
    int wid  = blockIdx.x * (blockDim.x >> 5) + (threadIdx.x >> 5);
    int lane = threadIdx.x & 31;
    int half = lane >> 4, lm = lane & 15;
    int mat = wid >> 6;
    int t   = wid & 63;
    int i0 = (t >> 3) << 4, j0 = (t & 7) << 4;
    const float* X = (mat < 4) ? (pred + mat * IMG) : (tgt + (mat - 4) * IMG);
    const float* C = cs + mat * IMG;
    float* D = (mat < 4) ? (ws + WS_PD + mat * IMG) : (ws + WS_TD + (mat - 4) * IMG);

    int kbase = j0 - 4; if (kbase < 0) kbase = 0;    // multiple of 4
    int row = i0 + lm;
    int j   = j0 + lm;
    v8f acc = {};
#pragma unroll
    for (int it = 0; it < 6; ++it) {
        int ka = kbase + 4 * it + 2 * half;          // even, may reach 130
        int kb = ka + 1;
        v2f a, b;
        b.x = (ka < HW) ? band9(ka, j) : 0.0f;       // zero B kills OOB rows
        b.y = (kb < HW) ? band9(kb, j) : 0.0f;
        int kc = ka < (HW - 2) ? ka : (HW - 2);      // pair load stays in row
        a.x = C[row * HW + kc];
        a.y = C[row * HW + kc + 1];
        acc = __builtin_amdgcn_wmma_f32_16x16x4_f32(false, a, false, b,
                                                    (short)0, acc, false, false);
    }
#pragma unroll
    for (int v = 0; v < 8; ++v) {
        int m = i0 + v + 8 * half;
        int n = j0 + lm;
        float den = rc9(m) * rc9(n);
        D[m * HW + n] = X[m * HW + n] - acc[v] / den;
    }
}

// ---- Kernel 3: per-block WMSE partial sums + min/max of dist arrays ----
__global__ void spl_stats(const float* __restrict__ pred,
                          const float* __restrict__ tgt,
                          float* __restrict__ ws) {
    __shared__ float sw[256], smn[256], smx[256];
    const float* pd = ws + WS_PD;
    const float* td = ws + WS_TD;
    int tid = threadIdx.x;
    int stride = gridDim.x * blockDim.x;
    float wsum = 0.0f, mn = 3.402823466e38f, mx = -3.402823466e38f;
    for (int i = blockIdx.x * blockDim.x + tid; i < NPIX; i += stride) {
        float p = pred[i], t = tgt[i];
        float w = 1.0f;
        if (t >= 0.5f)  w = 2.0f;
        if (t >= 2.0f)  w = 5.0f;
        if (t >= 5.0f)  w = 10.0f;
        if (t >= 10.0f) w = 30.0f;
        float d = p - t;
        wsum += w * d * d;
        float a = pd[i], b = td[i];
        mn = fminf(mn, fminf(a, b));
        mx = fmaxf(mx, fmaxf(a, b));
    }
    sw[tid] = wsum; smn[tid] = mn; smx[tid] = mx;
    __syncthreads();
    for (int s = 128; s > 0; s >>= 1) {
        if (tid < s) {
            sw[tid]  += sw[tid + s];
            smn[tid]  = fminf(smn[tid], smn[tid + s]);
            smx[tid]  = fmaxf(smx[tid], smx[tid + s]);
        }
        __syncthreads();
    }
    if (tid == 0) {
        ws[WS_WPART  + blockIdx.x] = sw[0];
        ws[WS_MNPART + blockIdx.x] = smn[0];
        ws[WS_MXPART + blockIdx.x] = smx[0];
    }
}

// ---- Kernel 4: reduce 64 partials -> lo, hi, wsum ----
__global__ void spl_reduce1(float* __restrict__ ws) {
    __shared__ float sw[64], smn[64], smx[64];
    int t = threadIdx.x;
    sw[t]  = ws[WS_WPART  + t];
    smn[t] = ws[WS_MNPART + t];
    smx[t] = ws[WS_MXPART + t];
    __syncthreads();
    for (int s = 32; s > 0; s >>= 1) {
        if (t < s) {
            sw[t]  += sw[t + s];
            smn[t]  = fminf(smn[t], smn[t + s]);
            smx[t]  = fmaxf(smx[t], smx[t + s]);
        }
        __syncthreads();
    }
    if (t == 0) {
        ws[WS_SCAL + 0] = smn[0];
        ws[WS_SCAL + 1] = smx[0];
        ws[WS_SCAL + 2] = sw[0];
    }
}

// number of k in [0,999] with (lo + k*dx) >= v   (matches grid comparisons)
__device__ __forceinline__ int cnt_ge(float v, float lo, float dx) {
    if (!(dx > 0.0f)) return (lo >= v) ? NINT : 0;
    float s = (v - lo) / dx;
    int k = (int)floorf(s) - 2;
    if (k < 0) k = 0;
    if (k > NINT) k = NINT;
    while (k < NINT && !((lo + (float)k * dx) >= v)) ++k;
    return NINT - k;
}

// ---- Kernel 5: per-block integer partials of |cnt(p)-cnt(t)| ----
__global__ void spl_crps(float* __restrict__ ws) {
    __shared__ int si[256];
    const float* pd = ws + WS_PD;
    const float* td = ws + WS_TD;
    int* ipart = (int*)(ws + WS_IPART);
    float lo = ws[WS_SCAL + 0];
    float hi = ws[WS_SCAL + 1];
    float dx = (hi - lo) / (float)(NINT - 1);
    int tid = threadIdx.x;
    int stride = gridDim.x * blockDim.x;
    int acc = 0;
    for (int i = blockIdx.x * blockDim.x + tid; i < NPIX; i += stride) {
        int cp = cnt_ge(pd[i], lo, dx);
        int ct = cnt_ge(td[i], lo, dx);
        int d = cp - ct; if (d < 0) d = -d;
        acc += d;
    }
    si[tid] = acc;
    __syncthreads();
    for (int s = 128; s > 0; s >>= 1) {
        if (tid < s) si[tid] += si[tid + s];
        __syncthreads();
    }
    if (tid == 0) ipart[blockIdx.x] = si[0];
}

// ---- Kernel 6: final scalar ----
__global__ void spl_final(const float* __restrict__ ws, float* __restrict__ out) {
    __shared__ int si[64];
    const int* ipart = (const int*)(ws + WS_IPART);
    int t = threadIdx.x;
    si[t] = ipart[t];
    __syncthreads();
    for (int s = 32; s > 0; s >>= 1) {
        if (t < s) si[t] += si[t + s];
        __syncthreads();
    }
    if (t == 0) {
        float lo = ws[WS_SCAL + 0];
        float hi = ws[WS_SCAL + 1];
        float dx = (hi - lo) / (float)(NINT - 1);
        float crps = dx * (float)si[0] / (float)NPIX;
        float wmse = ws[WS_SCAL + 2] / (float)NPIX;
        out[0] = 1e-4f * wmse + 1.0f * crps;
    }
}

extern "C" void kernel_launch(void* const* d_in, const int* in_sizes, int n_in,
                              void* d_out, int out_size, void* d_ws, size_t ws_size,
                              hipStream_t stream) {
    const float* pred = (const float*)d_in[0];
    const float* tgt  = (const float*)d_in[1];
    float* out = (float*)d_out;
    float* ws  = (float*)d_ws;
    float* cs  = ws + WS_CS;

    // 8 matrices x 64 tiles = 512 waves = 64 blocks of 8 waves (256 thr)
    spl_colsum<<<64, 256, 0, stream>>>(pred, tgt, cs);
    spl_rowsum_dist<<<64, 256, 0, stream>>>(pred, tgt, cs, ws);
    spl_stats<<<64, 256, 0, stream>>>(pred, tgt, ws);
    spl_reduce1<<<1, 64, 0, stream>>>(ws);
    spl_crps<<<64, 256, 0, stream>>>(ws);
    spl_final<<<1, 64, 0, stream>>>(ws, out);
}